// OccGridEmaBatched_21242908246605
// MI455X (gfx1250) — compile-verified
//
#include <hip/hip_runtime.h>
#include <hip/hip_bf16.h>
#include <stdint.h>

namespace {
constexpr int   kRes   = 160;
constexpr float kDecay = 0.95f;
constexpr float kThre  = 0.01f;
}

// ---------------------------------------------------------------------------
// K1: bulk grid copy through the gfx1250 async global<->LDS DMA path.
// Each lane moves one 16-byte chunk: global -> LDS slot -> global, tracked by
// ASYNCcnt (s_wait_asynccnt). Data never touches VGPRs.
// ---------------------------------------------------------------------------
__global__ void occ_copy_async_kernel(const float* __restrict__ src,
                                      float* __restrict__ dst,
                                      long long n4) {
  __shared__ __align__(16) float tile[256 * 4];
  long long i4 = (long long)blockIdx.x * blockDim.x + threadIdx.x;
  if (i4 >= n4) return;
  unsigned lds_off = (unsigned)(uintptr_t)(&tile[threadIdx.x * 4]);
  unsigned long long sa = (unsigned long long)(uintptr_t)src + (unsigned long long)i4 * 16ull;
  unsigned long long da = (unsigned long long)(uintptr_t)dst + (unsigned long long)i4 * 16ull;
  asm volatile(
      "global_load_async_to_lds_b128 %0, %1, off\n\t"
      "s_wait_asynccnt 0\n\t"
      "global_store_async_from_lds_b128 %2, %0, off\n\t"
      "s_wait_asynccnt 0\n\t"
      :
      : "v"(lds_off), "v"(sa), "v"(da)
      : "memory");
}

__device__ __forceinline__ int occ_lin_index(const float* __restrict__ pts,
                                             const int* __restrict__ bidx,
                                             int i) {
  float x = pts[3 * i + 0];
  float y = pts[3 * i + 1];
  float z = pts[3 * i + 2];
  // (pts/2 + 0.5) * res, truncate toward zero (matches astype(int32)), clamp
  int gx = (int)((x * 0.5f + 0.5f) * (float)kRes);
  int gy = (int)((y * 0.5f + 0.5f) * (float)kRes);
  int gz = (int)((z * 0.5f + 0.5f) * (float)kRes);
  gx = min(max(gx, 0), kRes - 1);
  gy = min(max(gy, 0), kRes - 1);
  gz = min(max(gz, 0), kRes - 1);
  int b = bidx[i];
  return ((b * kRes + gx) * kRes + gy) * kRes + gz;  // < 32,768,000, fits int
}

// ---------------------------------------------------------------------------
// K2: decay touched cells. All duplicate writers gather from the UNMODIFIED
// input grid, so they store bit-identical values -> benign deterministic race.
// ---------------------------------------------------------------------------
__global__ void occ_decay_scatter_kernel(const float* __restrict__ grid_in,
                                         const float* __restrict__ pts,
                                         const int* __restrict__ bidx,
                                         float* __restrict__ grid_out,
                                         int n) {
  int i = blockIdx.x * blockDim.x + threadIdx.x;
  if (i >= n) return;
  int lin = occ_lin_index(pts, bidx, i);
  grid_out[lin] = grid_in[lin] * kDecay;
}

// ---------------------------------------------------------------------------
// K3: scatter-max of val. Values are all >= 0, so uint-bitwise max == float
// max; lowers to native global_atomic_max_u32 resolved in L2 (grid is
// L2-resident: 131 MB < 192 MB).
// ---------------------------------------------------------------------------
__global__ void occ_max_scatter_kernel(const float* __restrict__ pts,
                                       const int* __restrict__ bidx,
                                       const float* __restrict__ val,
                                       float* __restrict__ grid_out,
                                       int n) {
  int i = blockIdx.x * blockDim.x + threadIdx.x;
  if (i >= n) return;
  int lin = occ_lin_index(pts, bidx, i);
  unsigned* out_u = reinterpret_cast<unsigned*>(grid_out);
  atomicMax(&out_u[lin], __float_as_uint(val[i]));
}

// ---------------------------------------------------------------------------
// K4: binarize: occ = grid > 0.01 (as 1.0f / 0.0f), vectorized 16B per lane.
// ---------------------------------------------------------------------------
__global__ void occ_threshold_kernel(const float4* __restrict__ grid,
                                     float4* __restrict__ occ,
                                     long long n4) {
  long long i = (long long)blockIdx.x * blockDim.x + threadIdx.x;
  if (i >= n4) return;
  float4 v = grid[i];
  float4 o;
  o.x = (v.x > kThre) ? 1.0f : 0.0f;
  o.y = (v.y > kThre) ? 1.0f : 0.0f;
  o.z = (v.z > kThre) ? 1.0f : 0.0f;
  o.w = (v.w > kThre) ? 1.0f : 0.0f;
  occ[i] = o;
}

extern "C" void kernel_launch(void* const* d_in, const int* in_sizes, int n_in,
                              void* d_out, int out_size, void* d_ws, size_t ws_size,
                              hipStream_t stream) {
  (void)n_in; (void)d_ws; (void)ws_size; (void)out_size;

  const float* grid_in = (const float*)d_in[0];  // [B,R,R,R] f32
  const float* pts     = (const float*)d_in[1];  // [N,3]     f32
  const int*   bidx    = (const int*)  d_in[2];  // [N]       i32
  const float* val     = (const float*)d_in[3];  // [N]       f32

  const long long ncell = (long long)in_sizes[0];   // 32,768,000
  const int       npts  = in_sizes[3];              // 4,194,304

  float* out_grid = (float*)d_out;            // first output: new_grid
  float* out_occ  = out_grid + ncell;         // second output: occ (as f32)

  const int  threads = 256;
  const long long n4 = ncell / 4;             // ncell divisible by 4
  const unsigned gblocks = (unsigned)((n4 + threads - 1) / threads);
  const unsigned pblocks = (unsigned)((npts + threads - 1) / threads);

  // 1) new_grid = occ_val_grid  (async DMA copy path)
  occ_copy_async_kernel<<<gblocks, threads, 0, stream>>>(grid_in, out_grid, n4);
  // 2) touched cells: new_grid[lin] = old[lin] * ema_decay
  occ_decay_scatter_kernel<<<pblocks, threads, 0, stream>>>(grid_in, pts, bidx,
                                                            out_grid, npts);
  // 3) touched cells: new_grid[lin] = max(new_grid[lin], val)
  occ_max_scatter_kernel<<<pblocks, threads, 0, stream>>>(pts, bidx, val,
                                                          out_grid, npts);
  // 4) occ = new_grid > occ_thre
  occ_threshold_kernel<<<gblocks, threads, 0, stream>>>(
      (const float4*)out_grid, (float4*)out_occ, n4);
}